// KNNGraph_21766894256201
// MI455X (gfx1250) — compile-verified
//
#include <hip/hip_runtime.h>

typedef __attribute__((ext_vector_type(2))) float v2f;
typedef __attribute__((ext_vector_type(8))) float v8f;

#define N_PTS 20000
#define DIM   128
#define KNN   16
#define NTILE (N_PTS / 16)     // 1250 column tiles, exact
#define WAVES 2                // waves per block, each covers NTILE/WAVES tiles
#define CT_PER_WAVE (NTILE / WAVES)  // 625, exact

#define FLT_BIG 3.4e38f

// ---------------------------------------------------------------------------
// Kernel 1: row squared-norms  sq[i] = sum_j x[i][j]^2
// ---------------------------------------------------------------------------
__global__ void knn_norms(const float* __restrict__ x, float* __restrict__ sq) {
    int i = blockIdx.x * blockDim.x + threadIdx.x;
    if (i < N_PTS) {
        const float4* xr = (const float4*)(x + (size_t)i * DIM);
        float s = 0.f;
#pragma unroll
        for (int j = 0; j < DIM / 4; ++j) {
            float4 v = xr[j];
            s += v.x * v.x + v.y * v.y + v.z * v.z + v.w * v.w;
        }
        sq[i] = s;
    }
}

// ---------------------------------------------------------------------------
// Kernel 2: fused (x @ x^T) + per-row top-16.
// Block = 2 waves; both waves own the same 16-row query tile but scan
// disjoint halves of the candidate set. f32 WMMA 16x16x4, 32 per col tile.
// ---------------------------------------------------------------------------
__global__ __launch_bounds__(32 * WAVES) void knn_main(const float* __restrict__ x,
                                                       const float* __restrict__ sq,
                                                       float* __restrict__ out) {
    __shared__ float tile[WAVES][16 * 17]; // per-wave d2 relayout buffer
    __shared__ float mD[16 * 64];          // 4 sorted 16-lists per row
    __shared__ int   mI[16 * 64];
    __shared__ float mD2[16 * 64];         // level-1 merged 32-lists
    __shared__ int   mI2[16 * 64];

    const int tid  = threadIdx.x;
    const int wv   = tid >> 5;          // wave id 0/1
    const int lane = tid & 31;
    const int hh   = lane >> 4;         // lane half
    const int lrow = lane & 15;
    const int row_base = blockIdx.x * 16;
    const int myrow    = row_base + lrow;

    // ---- preload A fragments (16 query rows x 128 depth) into VGPRs ----
    v2f A[32];
    {
        const float* arow = x + (size_t)myrow * DIM + hh * 2;
#pragma unroll
        for (int c = 0; c < 32; ++c)
            A[c] = *(const v2f*)(arow + c * 4);
    }

    float sqa[8];
#pragma unroll
    for (int r = 0; r < 8; ++r)
        sqa[r] = sq[row_base + r + 8 * hh];

    float bestD[16];
    int   bestI[16];
#pragma unroll
    for (int i = 0; i < 16; ++i) { bestD[i] = FLT_BIG; bestI[i] = -1; }

    const int ct0 = wv * CT_PER_WAVE;
    const int ct1 = ct0 + CT_PER_WAVE;

    for (int ct = ct0; ct < ct1; ++ct) {
        const int    col_base = ct * 16;
        const float* brow = x + (size_t)(col_base + lrow) * DIM + hh * 2;

        if (ct + 1 < ct1)
            __builtin_prefetch(brow + 16 * DIM, 0, 3);

        // ---- batch all 32 B fragments: 32 independent loads in flight ----
        v2f B[32];
#pragma unroll
        for (int c = 0; c < 32; ++c)
            B[c] = *(const v2f*)(brow + c * 4);

        // ---- two independent WMMA accumulation chains ----
        v8f a0 = {0.f, 0.f, 0.f, 0.f, 0.f, 0.f, 0.f, 0.f};
        v8f a1 = {0.f, 0.f, 0.f, 0.f, 0.f, 0.f, 0.f, 0.f};
#pragma unroll
        for (int c = 0; c < 32; c += 2) {
            a0 = __builtin_amdgcn_wmma_f32_16x16x4_f32(
                false, A[c],     false, B[c],     (short)0, a0, false, false);
            a1 = __builtin_amdgcn_wmma_f32_16x16x4_f32(
                false, A[c + 1], false, B[c + 1], (short)0, a1, false, false);
        }

        // ---- d2 = |q|^2 + |c|^2 - 2*dot ; relayout via per-wave LDS ----
        const float sqc = sq[col_base + lrow];
#pragma unroll
        for (int r = 0; r < 8; ++r) {
            float d2 = sqa[r] + sqc - 2.0f * (a0[r] + a1[r]);
            tile[wv][(r + 8 * hh) * 17 + lrow] = d2;
        }
        // intra-wave only: lanes are lockstep, DS ops in-order per wave;
        // fence+wave_barrier keep the compiler from reordering the DS ops.
        __builtin_amdgcn_fence(__ATOMIC_ACQ_REL, "wavefront");
        __builtin_amdgcn_wave_barrier();

        // ---- selection: lane handles row lrow, candidate half hh ----
#pragma unroll
        for (int j = 0; j < 8; ++j) {
            const int   cidx = col_base + hh * 8 + j;
            const float d    = tile[wv][lrow * 17 + hh * 8 + j];
            if (cidx != myrow && d < bestD[15]) {
                bestD[15] = d;
                bestI[15] = cidx;
#pragma unroll
                for (int i = 15; i > 0; --i) {
                    if (bestD[i] < bestD[i - 1]) {
                        float td = bestD[i]; bestD[i] = bestD[i - 1]; bestD[i - 1] = td;
                        int   ti = bestI[i]; bestI[i] = bestI[i - 1]; bestI[i - 1] = ti;
                    }
                }
            }
        }
        __builtin_amdgcn_fence(__ATOMIC_ACQ_REL, "wavefront");
        __builtin_amdgcn_wave_barrier();
    }

    // ---- publish 4 sorted partial lists per row: (wv, hh) -> slot ----
#pragma unroll
    for (int i = 0; i < 16; ++i) {
        mD[lrow * 64 + (wv * 2 + hh) * 16 + i] = bestD[i];
        mI[lrow * 64 + (wv * 2 + hh) * 16 + i] = bestI[i];
    }
    __syncthreads();

    // ---- level 1: merge list pairs (0,1) and (2,3) -> two sorted 32-lists ----
    if (tid < 32) {
        const int row = tid & 15;
        const int pr  = tid >> 4;               // which pair
        const float* dr = &mD[row * 64 + pr * 32];
        const int*   ir = &mI[row * 64 + pr * 32];
        float*       od = &mD2[row * 64 + pr * 32];
        int*         oi = &mI2[row * 64 + pr * 32];
        int p0 = 0, p1 = 16;
#pragma unroll
        for (int t = 0; t < 32; ++t) {
            float d0 = (p0 < 16) ? dr[p0] : FLT_BIG;
            float d1 = (p1 < 32) ? dr[p1] : FLT_BIG;
            bool take0 = d0 <= d1;
            od[t] = take0 ? d0 : d1;
            oi[t] = take0 ? ir[p0] : ir[p1];
            if (take0) ++p0; else ++p1;
        }
    }
    __syncthreads();

    // ---- level 2: merge the two 32-lists, emit first 16 indices ----
    if (tid < 16) {
        const int    row = tid;
        const float* dr  = &mD2[row * 64];
        const int*   ir  = &mI2[row * 64];
        float*       o   = out + (size_t)(row_base + row) * KNN;
        int p0 = 0, p1 = 32;
#pragma unroll
        for (int t = 0; t < KNN; ++t) {
            float d0 = (p0 < 32) ? dr[p0] : FLT_BIG;
            float d1 = (p1 < 64) ? dr[p1] : FLT_BIG;
            bool take0 = d0 <= d1;
            int  idx   = take0 ? ir[p0] : ir[p1];
            if (take0) ++p0; else ++p1;
            o[t] = (float)idx;
        }
    }
}

// ---------------------------------------------------------------------------
extern "C" void kernel_launch(void* const* d_in, const int* in_sizes, int n_in,
                              void* d_out, int out_size, void* d_ws, size_t ws_size,
                              hipStream_t stream) {
    const float* x  = (const float*)d_in[0];   // [N, D] float32
    // d_in[1] = k (==16), baked in as KNN
    float* out = (float*)d_out;                // [N, K] indices as floats
    float* sq  = (float*)d_ws;                 // [N] squared norms (80 KB scratch)

    knn_norms<<<(N_PTS + 255) / 256, 256, 0, stream>>>(x, sq);
    knn_main<<<NTILE, 32 * WAVES, 0, stream>>>(x, sq, out);
}